// RLMoERouter_43078521979512
// MI455X (gfx1250) — compile-verified
//
#include <hip/hip_runtime.h>
#include <hip/hip_bf16.h>
#include <stdint.h>

// RL MoE router (REINFORCE load-balancing). x (4,8192,2048) is shape-only in the
// reference -> we never read it (saves 256 MB of HBM traffic; problem becomes
// launch-latency bound). 2 tiny kernels:
//   K1: softmax CDF in LDS, 65536 categorical draws, write assignments+weights,
//       per-block 64-bin LDS histogram -> float partials in d_ws.
//   K2: one wave32. Reduce partial histograms with chained V_WMMA_F32_16X16X4_F32
//       (A = ones => result is K-layout invariant; counts < 2^24 => exact in f32),
//       then the O(64) reward / prefs / baseline update.

#define NUM_EXPERTS 64
#define KDRAWS      2
#define HIDDEN      2048
#define LR          0.1f

typedef float v2f __attribute__((ext_vector_type(2)));
typedef float v8f __attribute__((ext_vector_type(8)));

__device__ __forceinline__ uint32_t pcg_hash(uint32_t v) {
    uint32_t s = v * 747796405u + 2891336453u;
    uint32_t w = ((s >> ((s >> 28) + 4u)) ^ s) * 277803737u;
    return (w >> 22) ^ w;
}

// ---------------- Kernel 1: sample + per-block histogram ----------------
__global__ void rl_router_sample(const float* __restrict__ prefs,
                                 float* __restrict__ out_assign,   // [ndraws]
                                 float* __restrict__ out_weight,   // [ndraws]
                                 float* __restrict__ partial,      // [gridDim.x][64]
                                 int ndraws, int per_thread)
{
    __shared__ float    s_cdf [NUM_EXPERTS];
    __shared__ float    s_prob[NUM_EXPERTS];
    __shared__ unsigned s_hist[NUM_EXPERTS];

    const int t = threadIdx.x;
    if (t < NUM_EXPERTS) s_hist[t] = 0u;
    if (t == 0) {
        __builtin_prefetch(prefs, 0, 3);           // global_prefetch_b8
        // E=64 serial softmax + CDF: trivial cost, keeps it simple & exact.
        float mx = -3.402823466e38f;
        for (int e = 0; e < NUM_EXPERTS; ++e) mx = fmaxf(mx, prefs[e]);
        float sum = 0.f;
        for (int e = 0; e < NUM_EXPERTS; ++e) {
            float p = __expf(prefs[e] - mx);
            s_prob[e] = p; sum += p;
        }
        float inv = 1.f / sum, run = 0.f;
        for (int e = 0; e < NUM_EXPERTS; ++e) {
            s_prob[e] *= inv;
            run += s_prob[e];
            s_cdf[e] = run;
        }
        s_cdf[NUM_EXPERTS - 1] = 1.0f;             // guard against fp drift
    }
    __syncthreads();

    const int nthreads = (int)(gridDim.x * blockDim.x);
    const int g = (int)(blockIdx.x * blockDim.x + t);
    for (int j = 0; j < per_thread; ++j) {
        int i = j * nthreads + g;                  // coalesced draw index
        if (i < ndraws) {
            // Stateless counter-based RNG (reference uses jax threefry key 42;
            // exact bitstream is not reproducible here).
            uint32_t h = pcg_hash((uint32_t)i ^ 0x2a2a2a2au);
            float u = (float)(h >> 8) * (1.0f / 16777216.0f);   // [0,1)
            // 6-step branchless binary search over CDF in LDS
            int lo = 0;
            #pragma unroll
            for (int step = 32; step >= 1; step >>= 1)
                if (u >= s_cdf[lo + step - 1]) lo += step;
            lo = min(lo, NUM_EXPERTS - 1);
            out_assign[i] = (float)lo;
            out_weight[i] = s_prob[lo];
            atomicAdd(&s_hist[lo], 1u);            // ds_add_u32
        }
    }
    __syncthreads();
    if (t < NUM_EXPERTS)
        partial[blockIdx.x * NUM_EXPERTS + t] = (float)s_hist[t];
}

// ---------------- Kernel 2: WMMA histogram reduction + REINFORCE ----------------
__global__ void rl_router_finalize(const float* __restrict__ prefs,
                                   const float* __restrict__ partial, // [nblocks][64]
                                   float* __restrict__ out_prefs,     // [64]
                                   float* __restrict__ out_base,      // [1]
                                   int nblocks)
{
    __shared__ float s_cnt[NUM_EXPERTS];
    const int lane = (int)threadIdx.x;             // 32 lanes, all active

    // counts[n] = sum_k partial[k][n] via chained V_WMMA_F32_16X16X4_F32.
    // A = all-ones 16x4 => every row of D holds column sums of B, so the
    // result is invariant to the exact K<->lane mapping of the B operand.
    v2f a; a.x = 1.0f; a.y = 1.0f;
    const int n_in_grp = lane & 15;
    const int koff     = (lane >> 4) * 2;          // lanes 0-15: k+0/1, 16-31: k+2/3
    const int kmax     = nblocks - 1;
    for (int ng = 0; ng < NUM_EXPERTS; ng += 16) {
        v8f c = {};
        const int n = ng + n_in_grp;
        for (int kb = 0; kb < nblocks; kb += 4) {
            const int k0 = kb + koff, k1 = k0 + 1;
            // Clamp address (always in-bounds, unconditional load), then
            // select the value: branch-free v_cndmask, no EXEC churn.
            float x0 = partial[min(k0, kmax) * NUM_EXPERTS + n];
            float x1 = partial[min(k1, kmax) * NUM_EXPERTS + n];
            v2f b;
            b.x = (k0 < nblocks) ? x0 : 0.0f;
            b.y = (k1 < nblocks) ? x1 : 0.0f;
            c = __builtin_amdgcn_wmma_f32_16x16x4_f32(
                    false, a, false, b, (short)0, c, false, false);
        }
        if (lane < 16) s_cnt[ng + lane] = c[0];    // row 0 = column sums
    }
    __syncthreads();

    if (lane == 0) {
        float total = 0.f;
        for (int e = 0; e < NUM_EXPERTS; ++e) total += s_cnt[e];
        const float mean = total / (float)NUM_EXPERTS;
        float var = 0.f;
        for (int e = 0; e < NUM_EXPERTS; ++e) {
            float d = s_cnt[e] - mean;
            var += d * d;
        }
        var /= (float)(NUM_EXPERTS - 1);           // torch .std(): ddof=1
        const float reward = -(sqrtf(var) / mean);
        const float adv    = reward - 0.0f;        // baseline starts at 0
        // softmax(prefs), two passes (no private array -> no scratch)
        float mx = -3.402823466e38f;
        for (int e = 0; e < NUM_EXPERTS; ++e) mx = fmaxf(mx, prefs[e]);
        float sum = 0.f;
        for (int e = 0; e < NUM_EXPERTS; ++e) sum += __expf(prefs[e] - mx);
        const float inv    = 1.f / sum;
        const float invtot = 1.f / total;
        for (int e = 0; e < NUM_EXPERTS; ++e) {
            float prob = __expf(prefs[e] - mx) * inv;
            float freq = s_cnt[e] * invtot;
            out_prefs[e] = prefs[e] + LR * adv * (freq - prob);
        }
        out_base[0] = LR * adv;                    // 0 + LR*adv
    }
}

extern "C" void kernel_launch(void* const* d_in, const int* in_sizes, int n_in,
                              void* d_out, int out_size, void* d_ws, size_t ws_size,
                              hipStream_t stream) {
    const float* prefs = (const float*)d_in[1];
    // x is shape-only: tokens = B*S = numel(x)/HIDDEN; draws = tokens*K
    const int tokens = in_sizes[0] / HIDDEN;       // 32768
    const int ndraws = tokens * KDRAWS;            // 65536

    const int threads = 256;
    int blocks = (ndraws + threads * 4 - 1) / (threads * 4);   // 64
    // keep partial histograms inside d_ws
    int max_blocks = (int)(ws_size / (NUM_EXPERTS * sizeof(float)));
    if (max_blocks < 1) max_blocks = 1;
    if (blocks > max_blocks) blocks = max_blocks;
    if (blocks >= 4) blocks &= ~3;                 // multiple of 4 -> selects are no-ops
    const int per_thread = (ndraws + blocks * threads - 1) / (blocks * threads);

    float* out       = (float*)d_out;
    float* o_assign  = out;                        // [ndraws]
    float* o_weight  = out + ndraws;               // [ndraws]
    float* o_prefs   = out + 2 * ndraws;           // [64]
    float* o_base    = out + 2 * ndraws + NUM_EXPERTS; // [1]
    float* partial   = (float*)d_ws;               // [blocks][64]

    rl_router_sample<<<blocks, threads, 0, stream>>>(
        prefs, o_assign, o_weight, partial, ndraws, per_thread);
    rl_router_finalize<<<1, 32, 0, stream>>>(
        prefs, partial, o_prefs, o_base, blocks);
}